// CoSaMP_56573309224253
// MI455X (gfx1250) — compile-verified
//
#include <hip/hip_runtime.h>
#include <math.h>

#define Mdim  1024
#define Ndim  4096
#define Bdim  16
#define Ksp   32
#define SMAXC 512
#define ITERS 8
#define TOLER 1e-6f

typedef __attribute__((ext_vector_type(2))) float v2f;
typedef __attribute__((ext_vector_type(8))) float v8f;

__device__ __forceinline__ v8f wmma_f32(v2f a, v2f b, v8f c) {
  // D = A(16x4 f32) * B(4x16 f32) + C(16x16 f32), exact fp32 on the matrix pipe
  return __builtin_amdgcn_wmma_f32_16x16x4_f32(false, a, false, b, (short)0, c,
                                               false, false);
}

// ---------------------------------------------------------------------------
// init: residual = y, residualT = y^T, support = 0, done = 0
// ---------------------------------------------------------------------------
__global__ void cosamp_init(const float* __restrict__ meas,
                            float* __restrict__ resid,
                            float* __restrict__ residT,
                            int* __restrict__ support,
                            int* __restrict__ done) {
  int g = blockIdx.x * blockDim.x + threadIdx.x;
  if (g < Bdim * Ndim) support[g] = 0;
  if (g < Bdim * Mdim) {
    int b = g / Mdim, m = g % Mdim;
    float v = meas[g];
    resid[g] = v;
    residT[m * Bdim + b] = v;
  }
  if (g < Bdim) done[g] = 0;
}

// ---------------------------------------------------------------------------
// proxy[b][n] = sum_m A[m][n] * resid[b][m]   via WMMA f32 16x16x4
// One wave per 16-row tile of N; the 16 batches are the 16 output columns.
// ---------------------------------------------------------------------------
__global__ void cosamp_proxy(const float* __restrict__ A,
                             const float* __restrict__ residT,
                             float* __restrict__ proxy) {
  int n0 = blockIdx.x * 16;
  int lane = threadIdx.x;
  int sp = lane & 15;
  int khalf = (lane >> 4) << 1;  // lanes 0-15 -> K 0,1 ; lanes 16-31 -> K 2,3
  v8f acc = {};
  for (int m = 0; m < Mdim; m += 4) {
    int klo = m + khalf;
    v2f a, bb;
    a.x = A[klo * Ndim + n0 + sp];          // A^T[n][m] = A[m][n]
    a.y = A[(klo + 1) * Ndim + n0 + sp];
    bb.x = residT[klo * Bdim + sp];         // R^T[m][b]
    bb.y = residT[(klo + 1) * Bdim + sp];
    acc = wmma_f32(a, bb, acc);
  }
  int rhi = (lane >> 4) * 8;
  for (int r = 0; r < 8; ++r)
    proxy[sp * Ndim + n0 + r + rhi] = acc[r];
}

// ---------------------------------------------------------------------------
// top-64 of |proxy| per batch -> support; compact support indices (ascending)
// into idx[0..SMAX) with validf mask (padding = first non-support indices).
// ---------------------------------------------------------------------------
__global__ void cosamp_topk(const float* __restrict__ proxy,
                            int* __restrict__ support,
                            int* __restrict__ idxA,
                            float* __restrict__ validf,
                            const int* __restrict__ done) {
  int b = blockIdx.x;
  if (done[b]) return;
  int tid = threadIdx.x;
  __shared__ float vals[Ndim];
  __shared__ float redV[256];
  __shared__ int redI[256];
  __shared__ int scan[256];

  for (int n = tid; n < Ndim; n += 256) vals[n] = fabsf(proxy[b * Ndim + n]);
  __syncthreads();

  int* supB = support + b * Ndim;
  for (int round = 0; round < 2 * Ksp; ++round) {
    float best = -1.f; int bi = 0;
    for (int n = tid; n < Ndim; n += 256) {
      float v = vals[n];
      if (v > best) { best = v; bi = n; }
    }
    redV[tid] = best; redI[tid] = bi;
    __syncthreads();
    for (int s = 128; s > 0; s >>= 1) {
      if (tid < s) {
        if (redV[tid + s] > redV[tid] ||
            (redV[tid + s] == redV[tid] && redI[tid + s] < redI[tid])) {
          redV[tid] = redV[tid + s]; redI[tid] = redI[tid + s];
        }
      }
      __syncthreads();
    }
    if (tid == 0) { int w = redI[0]; supB[w] = 1; vals[w] = -2.f; }
    __syncthreads();
  }
  __threadfence_block();
  __syncthreads();

  // compaction: stable ascending order, support first then padding
  int base = tid * 16;
  int c = 0;
  for (int q = 0; q < 16; ++q) c += supB[base + q];
  scan[tid] = c;
  __syncthreads();
  for (int off = 1; off < 256; off <<= 1) {
    int addv = (tid >= off) ? scan[tid - off] : 0;
    __syncthreads();
    scan[tid] += addv;
    __syncthreads();
  }
  int total = scan[255];
  int excl = scan[tid] - c;
  int supPos = excl;
  int nonPos = base - excl;
  int* idxB = idxA + b * SMAXC;
  for (int q = 0; q < 16; ++q) {
    int g = base + q;
    if (supB[g]) {
      idxB[supPos++] = g;
    } else {
      if (total + nonPos < SMAXC) idxB[total + nonPos] = g;
      nonPos++;
    }
  }
  for (int s = tid; s < SMAXC; s += 256)
    validf[b * SMAXC + s] = (s < total) ? 1.f : 0.f;
}

// ---------------------------------------------------------------------------
// rhs[s] = valid[s] * sum_m A[m][idx[s]] * resid[b][m]
// ---------------------------------------------------------------------------
__global__ void cosamp_rhs(const float* __restrict__ A,
                           const float* __restrict__ resid,
                           const int* __restrict__ idxA,
                           const float* __restrict__ validf,
                           float* __restrict__ rhs,
                           const int* __restrict__ done) {
  int b = blockIdx.x;
  if (done[b]) return;
  int tid = threadIdx.x;
  const float* rb = resid + b * Mdim;
  for (int s = tid; s < SMAXC; s += 256) {
    int col = idxA[b * SMAXC + s];
    float v = validf[b * SMAXC + s];
    float sum = 0.f;
    for (int m = 0; m < Mdim; ++m) sum += A[m * Ndim + col] * rb[m];
    rhs[b * SMAXC + s] = sum * v;
  }
}

// ---------------------------------------------------------------------------
// G = A_sub^T A_sub + diag(1 - valid)   via WMMA f32, 64x64 macro-tiles.
// grid (8,8,B), 256 threads = 8 waves; each wave owns two 16x16 subtiles.
// ---------------------------------------------------------------------------
__global__ void cosamp_gram(const float* __restrict__ A,
                            const int* __restrict__ idxA,
                            const float* __restrict__ validf,
                            float* __restrict__ G,
                            const int* __restrict__ done) {
  int b = blockIdx.z;
  if (done[b]) return;
  const int* idxB = idxA + b * SMAXC;
  const float* vldB = validf + b * SMAXC;
  float* Gb = G + (size_t)b * SMAXC * SMAXC;
  int i0 = blockIdx.x * 64;
  int j0 = blockIdx.y * 64;
  __shared__ float lAi[32 * 64];
  __shared__ float lAj[32 * 64];
  int tid = threadIdx.x;
  int lane = tid & 31;
  int w = tid >> 5;
  int si = w >> 2;   // 0..1 (wave also covers si+2)
  int sj = w & 3;    // 0..3
  int sp = lane & 15;
  int khalf = (lane >> 4) << 1;
  v8f acc0 = {}; v8f acc1 = {};
  for (int m0 = 0; m0 < Mdim; m0 += 32) {
    for (int e = tid; e < 32 * 64; e += 256) {
      int row = e >> 6, col = e & 63;
      lAi[e] = A[(m0 + row) * Ndim + idxB[i0 + col]] * vldB[i0 + col];
      lAj[e] = A[(m0 + row) * Ndim + idxB[j0 + col]] * vldB[j0 + col];
      if (m0 + 32 < Mdim)
        __builtin_prefetch(&A[(m0 + 32 + row) * Ndim + idxB[i0 + col]], 0, 1);
    }
    __syncthreads();
    for (int mk = 0; mk < 32; mk += 4) {
      int klo = mk + khalf;
      v2f bfrag, a0, a1;
      bfrag.x = lAj[klo * 64 + sj * 16 + sp];
      bfrag.y = lAj[(klo + 1) * 64 + sj * 16 + sp];
      a0.x = lAi[klo * 64 + si * 16 + sp];
      a0.y = lAi[(klo + 1) * 64 + si * 16 + sp];
      a1.x = lAi[klo * 64 + (si + 2) * 16 + sp];
      a1.y = lAi[(klo + 1) * 64 + (si + 2) * 16 + sp];
      acc0 = wmma_f32(a0, bfrag, acc0);
      acc1 = wmma_f32(a1, bfrag, acc1);
    }
    __syncthreads();
  }
  int rhi = (lane >> 4) * 8;
  for (int r = 0; r < 8; ++r) {
    int gj = j0 + sj * 16 + sp;
    int gi = i0 + si * 16 + r + rhi;
    float v = acc0[r];
    if (gi == gj) v += (vldB[gi] > 0.5f) ? 0.f : 1.f;
    Gb[gi * SMAXC + gj] = v;
    gi = i0 + (si + 2) * 16 + r + rhi;
    v = acc1[r];
    if (gi == gj) v += (vldB[gi] > 0.5f) ? 0.f : 1.f;
    Gb[gi * SMAXC + gj] = v;
  }
}

// ---------------------------------------------------------------------------
// In-place Cholesky of G (lower triangle), then forward/back substitution.
// One block per batch.
// ---------------------------------------------------------------------------
__global__ void cosamp_chol(float* __restrict__ G,
                            const float* __restrict__ rhs,
                            const float* __restrict__ validf,
                            float* __restrict__ sol,
                            const int* __restrict__ done) {
  int b = blockIdx.x;
  if (done[b]) return;
  int tid = threadIdx.x;
  float* Gb = G + (size_t)b * SMAXC * SMAXC;
  __shared__ float red[256];
  __shared__ float zsh[SMAXC];

  for (int j = 0; j < SMAXC; ++j) {
    float p = 0.f;
    for (int t = tid; t < j; t += 256) {
      float l = Gb[j * SMAXC + t];
      p += l * l;
    }
    red[tid] = p;
    __syncthreads();
    for (int s = 128; s > 0; s >>= 1) {
      if (tid < s) red[tid] += red[tid + s];
      __syncthreads();
    }
    if (tid == 0) {
      float d = Gb[j * SMAXC + j] - red[0];
      Gb[j * SMAXC + j] = sqrtf(fmaxf(d, 1e-20f));
    }
    __syncthreads();
    float Ljj = Gb[j * SMAXC + j];
    for (int i = j + 1 + tid; i < SMAXC; i += 256) {
      float dot = 0.f;
      for (int p2 = 0; p2 < j; ++p2)
        dot += Gb[i * SMAXC + p2] * Gb[j * SMAXC + p2];
      Gb[i * SMAXC + j] = (Gb[i * SMAXC + j] - dot) / Ljj;
    }
    __syncthreads();
  }

  // triangular solves on wave 0 (shfl reductions, no barriers needed)
  if (tid < 32) {
    for (int i = 0; i < SMAXC; ++i) {           // forward: L z = rhs
      float p = 0.f;
      for (int t = tid; t < i; t += 32) p += Gb[i * SMAXC + t] * zsh[t];
      for (int off = 16; off > 0; off >>= 1) p += __shfl_xor(p, off, 32);
      if (tid == 0)
        zsh[i] = (rhs[b * SMAXC + i] - p) / Gb[i * SMAXC + i];
    }
    for (int i = SMAXC - 1; i >= 0; --i) {      // backward: L^T x = z (in place)
      float p = 0.f;
      for (int t = i + 1 + tid; t < SMAXC; t += 32) p += Gb[t * SMAXC + i] * zsh[t];
      for (int off = 16; off > 0; off >>= 1) p += __shfl_xor(p, off, 32);
      if (tid == 0)
        zsh[i] = (zsh[i] - p) / Gb[i * SMAXC + i];
    }
  }
  __syncthreads();
  for (int s = tid; s < SMAXC; s += 256)
    sol[b * SMAXC + s] = zsh[s] * validf[b * SMAXC + s];
}

// ---------------------------------------------------------------------------
// Hard threshold to top-32, write x, recompute residual, convergence check.
// ---------------------------------------------------------------------------
__global__ void cosamp_thresh(const float* __restrict__ A,
                              const float* __restrict__ meas,
                              const int* __restrict__ idxA,
                              const float* __restrict__ sol,
                              float* __restrict__ resid,
                              float* __restrict__ residT,
                              float* __restrict__ xout,
                              int* __restrict__ done) {
  int b = blockIdx.x;
  if (done[b]) return;
  int tid = threadIdx.x;
  __shared__ float sv[SMAXC];
  __shared__ float av[SMAXC];
  __shared__ float redV[256];
  __shared__ int redI[256];
  __shared__ float wval[Ksp];
  __shared__ int wcol[Ksp];

  for (int s = tid; s < SMAXC; s += 256) {
    float v = sol[b * SMAXC + s];
    sv[s] = v;
    av[s] = fabsf(v);
  }
  for (int n = tid; n < Ndim; n += 256) xout[b * Ndim + n] = 0.f;
  __syncthreads();

  for (int r = 0; r < Ksp; ++r) {
    float best = -1.f; int bi = 0;
    for (int s = tid; s < SMAXC; s += 256) {
      if (av[s] > best) { best = av[s]; bi = s; }
    }
    redV[tid] = best; redI[tid] = bi;
    __syncthreads();
    for (int s = 128; s > 0; s >>= 1) {
      if (tid < s) {
        if (redV[tid + s] > redV[tid] ||
            (redV[tid + s] == redV[tid] && redI[tid + s] < redI[tid])) {
          redV[tid] = redV[tid + s]; redI[tid] = redI[tid + s];
        }
      }
      __syncthreads();
    }
    if (tid == 0) {
      int slot = redI[0];
      wval[r] = sv[slot];
      wcol[r] = idxA[b * SMAXC + slot];
      av[slot] = -2.f;
    }
    __syncthreads();
  }

  __threadfence_block();
  __syncthreads();
  if (tid < Ksp) xout[b * Ndim + wcol[tid]] = wval[tid];

  // r_new = y - A x_new  (x_new is 32-sparse)
  float nrm = 0.f;
  for (int m = tid; m < Mdim; m += 256) {
    float r = meas[b * Mdim + m];
    for (int t = 0; t < Ksp; ++t) r -= wval[t] * A[m * Ndim + wcol[t]];
    resid[b * Mdim + m] = r;
    residT[m * Bdim + b] = r;
    nrm += r * r;
  }
  redV[tid] = nrm;
  __syncthreads();
  for (int s = 128; s > 0; s >>= 1) {
    if (tid < s) redV[tid] += redV[tid + s];
    __syncthreads();
  }
  if (tid == 0 && redV[0] < TOLER * TOLER) done[b] = 1;
}

// ---------------------------------------------------------------------------
extern "C" void kernel_launch(void* const* d_in, const int* in_sizes, int n_in,
                              void* d_out, int out_size, void* d_ws, size_t ws_size,
                              hipStream_t stream) {
  const float* meas = (const float*)d_in[0];  // [B, M]
  const float* A    = (const float*)d_in[1];  // [M, N]
  float* xout = (float*)d_out;                // [B, N]
  char* ws = (char*)d_ws;

  size_t off = 0;
  float* resid  = (float*)(ws + off); off += (size_t)Bdim * Mdim * 4;
  float* residT = (float*)(ws + off); off += (size_t)Mdim * Bdim * 4;
  float* proxy  = (float*)(ws + off); off += (size_t)Bdim * Ndim * 4;
  int*   support= (int*)  (ws + off); off += (size_t)Bdim * Ndim * 4;
  int*   idxA   = (int*)  (ws + off); off += (size_t)Bdim * SMAXC * 4;
  float* validf = (float*)(ws + off); off += (size_t)Bdim * SMAXC * 4;
  float* rhs    = (float*)(ws + off); off += (size_t)Bdim * SMAXC * 4;
  float* sol    = (float*)(ws + off); off += (size_t)Bdim * SMAXC * 4;
  int*   done   = (int*)  (ws + off); off += 256;
  float* G      = (float*)(ws + off); off += (size_t)Bdim * SMAXC * SMAXC * 4;

  cosamp_init<<<(Bdim * Ndim + 255) / 256, 256, 0, stream>>>(meas, resid, residT,
                                                             support, done);
  for (int it = 0; it < ITERS; ++it) {
    cosamp_proxy<<<Ndim / 16, 32, 0, stream>>>(A, residT, proxy);
    cosamp_topk<<<Bdim, 256, 0, stream>>>(proxy, support, idxA, validf, done);
    cosamp_rhs<<<Bdim, 256, 0, stream>>>(A, resid, idxA, validf, rhs, done);
    cosamp_gram<<<dim3(SMAXC / 64, SMAXC / 64, Bdim), 256, 0, stream>>>(
        A, idxA, validf, G, done);
    cosamp_chol<<<Bdim, 256, 0, stream>>>(G, rhs, validf, sol, done);
    cosamp_thresh<<<Bdim, 256, 0, stream>>>(A, meas, idxA, sol, resid, residT,
                                            xout, done);
  }
}